// DeltaNetLayer_65652870087354
// MI455X (gfx1250) — compile-verified
//
#include <hip/hip_runtime.h>
#include <hip/hip_bf16.h>

typedef __attribute__((ext_vector_type(2))) float v2f;
typedef __attribute__((ext_vector_type(8))) float v8f;

#define D_DIM 256
#define L_DIM 1024
#define B_DIM 4
#define M_DIM (B_DIM * L_DIM)   // 4096 token rows
#define LN_EPS 1e-5f

// ---------------------------------------------------------------------------
// Shared-memory block reduction (256 threads = 8 wave32). sh8 = 8 floats.
// ---------------------------------------------------------------------------
__device__ __forceinline__ float block_sum_256(float v, float* sh8) {
#pragma unroll
  for (int m = 16; m > 0; m >>= 1) v += __shfl_xor(v, m, 32);
  if ((threadIdx.x & 31) == 0) sh8[threadIdx.x >> 5] = v;
  __syncthreads();
  float r = 0.f;
#pragma unroll
  for (int i = 0; i < 8; ++i) r += sh8[i];
  return r;
}

// ---------------------------------------------------------------------------
// f32 WMMA GEMM: Out[M,256] = A[M,256] @ Wm[256,256]^T (+ bias)
// One 16x16 output tile per wave via V_WMMA_F32_16X16X4_F32 (K-loop of 64).
// Fragment layouts per CDNA5 ISA 7.12.2 (32-bit A 16x4, B 4x16, C/D 16x16).
// Grid: (M/128, 256/16), block 256 (8 waves).
// ---------------------------------------------------------------------------
__global__ void wmma_gemm_f32_kernel(const float* __restrict__ A,
                                     const float* __restrict__ Wm,
                                     float* __restrict__ Out,
                                     const float* __restrict__ bias) {
  const int K = D_DIM, N = D_DIM;
  const int wave = threadIdx.x >> 5;
  const int lane = threadIdx.x & 31;
  const int half = lane >> 4;        // 0: K+0..1, 1: K+2..3
  const int l16  = lane & 15;
  const int tM = (blockIdx.x * 8 + wave) * 16;
  const int tN = blockIdx.y * 16;

  // A fragment source: row m = tM + l16, two consecutive K values per step.
  const float* arow = A  + (size_t)(tM + l16) * K + (half ? 2 : 0);
  // B fragment: B[k][n] = Wm[n][k]  (Out = A @ Wm^T), n = tN + l16.
  const float* brow = Wm + (size_t)(tN + l16) * K + (half ? 2 : 0);

  v8f acc = {};
#pragma unroll 8
  for (int kc = 0; kc < K; kc += 4) {
    v2f a; a.x = arow[kc]; a.y = arow[kc + 1];
    v2f b; b.x = brow[kc]; b.y = brow[kc + 1];
    acc = __builtin_amdgcn_wmma_f32_16x16x4_f32(
        /*neg_a=*/false, a, /*neg_b=*/false, b,
        /*c_mod=*/(short)0, acc, /*reuse_a=*/false, /*reuse_b=*/false);
  }

  const int n = tN + l16;
  const float bv = bias ? bias[n] : 0.f;
  union { v8f v; float f[8]; } u; u.v = acc;
#pragma unroll
  for (int r = 0; r < 8; ++r) {
    const int m = tM + r + (half ? 8 : 0);   // C/D: VGPR r -> M=r (+8 for hi lanes)
    Out[(size_t)m * N + n] = u.f[r] + bv;
  }
}

// ---------------------------------------------------------------------------
// Pointwise: q,k (in place) -> phi = LN(elu(.)+1) ; beta = sigmoid(x.bw + bb)
// One block (256 threads) per token row.
// ---------------------------------------------------------------------------
__global__ void phi_beta_kernel(const float* __restrict__ x,
                                float* __restrict__ q,
                                float* __restrict__ k,
                                const float* __restrict__ beta_w,
                                const float* __restrict__ beta_b,
                                const float* __restrict__ lnp_g,
                                const float* __restrict__ lnp_b,
                                float* __restrict__ beta) {
  __shared__ float sh[8 * 5];
  const int row = blockIdx.x, tid = threadIdx.x;
  const size_t off = (size_t)row * D_DIM + tid;

  float qe; { float t = q[off]; qe = (t > 0.f) ? t + 1.f : __expf(t); }
  float ke; { float t = k[off]; ke = (t > 0.f) ? t + 1.f : __expf(t); }
  const float xb = x[off] * beta_w[tid];

  const float sq  = block_sum_256(qe,      sh + 0);
  const float sq2 = block_sum_256(qe * qe, sh + 8);
  const float sk  = block_sum_256(ke,      sh + 16);
  const float sk2 = block_sum_256(ke * ke, sh + 24);
  const float sb  = block_sum_256(xb,      sh + 32);

  const float muq = sq / D_DIM, varq = sq2 / D_DIM - muq * muq;
  const float muk = sk / D_DIM, vark = sk2 / D_DIM - muk * muk;
  const float g = lnp_g[tid], bb = lnp_b[tid];
  q[off] = (qe - muq) * rsqrtf(varq + LN_EPS) * g + bb;
  k[off] = (ke - muk) * rsqrtf(vark + LN_EPS) * g + bb;
  if (tid == 0) beta[row] = 1.f / (1.f + __expf(-(sb + beta_b[0])));
}

// ---------------------------------------------------------------------------
// Row-parallel fast-weight scan. One wave per (batch, row): the delta-rule
// recurrence is row-separable, so each wave evolves one 256-float row of W
// held entirely in 8 VGPRs/lane. Per step: two 8-FMA partials, two wave32
// xor-reductions, rank-1 register update. Next-step phi loads are issued
// before the compute (software pipelining); far-ahead __builtin_prefetch
// keeps the phi streams hot (they are L2-resident: 8 MB << 192 MB L2).
// Grid: (32, B), block 256 (8 rows per block).
// ---------------------------------------------------------------------------
union F8 { float4 v4[2]; float f[8]; };

__global__ void delta_scan_kernel(const float* __restrict__ phi_q,
                                  const float* __restrict__ phi_k,
                                  const float* __restrict__ v,
                                  const float* __restrict__ beta,
                                  const float* __restrict__ decay,
                                  float* __restrict__ y) {
  const int b    = blockIdx.y;
  const int wave = threadIdx.x >> 5;
  const int lane = threadIdx.x & 31;
  const int row  = blockIdx.x * 8 + wave;   // 0..255
  const int c0   = lane * 8;                // this lane's 8 columns

  const float df = 1.f / (1.f + __expf(-decay[0]));
  const size_t base = (size_t)b * L_DIM * D_DIM;

  float w[8];
#pragma unroll
  for (int c = 0; c < 8; ++c) w[c] = 0.f;

  // preload t = 0
  F8 cq, ck;
  cq.v4[0] = *(const float4*)(phi_q + base + c0);
  cq.v4[1] = *(const float4*)(phi_q + base + c0 + 4);
  ck.v4[0] = *(const float4*)(phi_k + base + c0);
  ck.v4[1] = *(const float4*)(phi_k + base + c0 + 4);
  float cv = v[base + row];
  float cb = beta[(size_t)b * L_DIM];

  for (int t = 0; t < L_DIM; ++t) {
    // ---- issue next-step loads first (no dependency on current compute) ----
    const int tn = (t + 1 < L_DIM) ? t + 1 : t;
    const size_t noff = base + (size_t)tn * D_DIM;
    F8 nq, nk;
    nq.v4[0] = *(const float4*)(phi_q + noff + c0);
    nq.v4[1] = *(const float4*)(phi_q + noff + c0 + 4);
    nk.v4[0] = *(const float4*)(phi_k + noff + c0);
    nk.v4[1] = *(const float4*)(phi_k + noff + c0 + 4);
    const float nv = v[noff + row];
    const float nb = beta[(size_t)b * L_DIM + tn];
    if (t + 8 < L_DIM) {
      const size_t poff = base + (size_t)(t + 8) * D_DIM + c0;
      __builtin_prefetch(phi_q + poff, 0, 0);   // global_prefetch_b8
      __builtin_prefetch(phi_k + poff, 0, 0);
    }

    // ---- retrieved = W.phi_q ; v_bar = W.phi_k (pre-update W) ----
    float rq = 0.f, rk = 0.f;
#pragma unroll
    for (int c = 0; c < 8; ++c) {
      rq = fmaf(w[c], cq.f[c], rq);
      rk = fmaf(w[c], ck.f[c], rk);
    }
#pragma unroll
    for (int m = 16; m > 0; m >>= 1) {
      rq += __shfl_xor(rq, m, 32);
      rk += __shfl_xor(rk, m, 32);
    }
    if (lane == 0) y[base + (size_t)t * D_DIM + row] = rq;

    // ---- W <- df*W + (beta*(v - v_bar)) outer phi_k ----
    const float u = cb * (cv - rk);
#pragma unroll
    for (int c = 0; c < 8; ++c) w[c] = fmaf(df, w[c], u * ck.f[c]);

    cq = nq; ck = nk; cv = nv; cb = nb;
  }
}

// ---------------------------------------------------------------------------
// Row layernorm of y in place. One block (256 threads) per token row.
// ---------------------------------------------------------------------------
__global__ void ln_rows_kernel(float* __restrict__ y,
                               const float* __restrict__ g,
                               const float* __restrict__ b) {
  __shared__ float sh[16];
  const int row = blockIdx.x, tid = threadIdx.x;
  const size_t off = (size_t)row * D_DIM + tid;
  const float v  = y[off];
  const float s  = block_sum_256(v, sh);
  const float s2 = block_sum_256(v * v, sh + 8);
  const float mu = s / D_DIM, var = s2 / D_DIM - mu * mu;
  y[off] = (v - mu) * rsqrtf(var + LN_EPS) * g[tid] + b[tid];
}

// ---------------------------------------------------------------------------
extern "C" void kernel_launch(void* const* d_in, const int* in_sizes, int n_in,
                              void* d_out, int out_size, void* d_ws, size_t ws_size,
                              hipStream_t stream) {
  (void)in_sizes; (void)n_in; (void)out_size; (void)ws_size;
  const float* x      = (const float*)d_in[0];
  const float* Wq     = (const float*)d_in[1];
  const float* Wk     = (const float*)d_in[2];
  const float* Wv     = (const float*)d_in[3];
  const float* beta_w = (const float*)d_in[4];
  const float* beta_b = (const float*)d_in[5];
  const float* decay  = (const float*)d_in[6];
  const float* Wo     = (const float*)d_in[7];
  const float* bo     = (const float*)d_in[8];
  const float* ln_g   = (const float*)d_in[9];
  const float* ln_b   = (const float*)d_in[10];
  const float* lnp_g  = (const float*)d_in[11];
  const float* lnp_b  = (const float*)d_in[12];
  float* out = (float*)d_out;

  // workspace layout (floats): q(1M) k(1M) v(1M) beta(4096 + pad) y(1M) ~ 16.8 MB
  char* ws = (char*)d_ws;
  const size_t MAT = (size_t)M_DIM * D_DIM * sizeof(float);   // 4 MB
  float* q    = (float*)(ws);
  float* k    = (float*)(ws + MAT);
  float* vv   = (float*)(ws + 2 * MAT);
  float* beta = (float*)(ws + 3 * MAT);
  float* y    = (float*)(ws + 3 * MAT + (64 << 10));

  const dim3 gemm_grid(M_DIM / 128, D_DIM / 16);   // (32, 16)
  const dim3 gemm_blk(256);

  // 1) projections (f32 WMMA)
  wmma_gemm_f32_kernel<<<gemm_grid, gemm_blk, 0, stream>>>(x, Wq, q, nullptr);
  wmma_gemm_f32_kernel<<<gemm_grid, gemm_blk, 0, stream>>>(x, Wk, k, nullptr);
  wmma_gemm_f32_kernel<<<gemm_grid, gemm_blk, 0, stream>>>(x, Wv, vv, nullptr);

  // 2) phi_q/phi_k (in place over q/k) + beta
  phi_beta_kernel<<<dim3(M_DIM), dim3(256), 0, stream>>>(
      x, q, k, beta_w, beta_b, lnp_g, lnp_b, beta);

  // 3) row-parallel fast-weight scan (wave per (batch,row), state in VGPRs)
  delta_scan_kernel<<<dim3(32, B_DIM), dim3(256), 0, stream>>>(
      q, k, vv, beta, decay, y);

  // 4) layernorm of y (in place)
  ln_rows_kernel<<<dim3(M_DIM), dim3(256), 0, stream>>>(y, ln_g, ln_b);

  // 5) output projection + bias (f32 WMMA)
  wmma_gemm_f32_kernel<<<gemm_grid, gemm_blk, 0, stream>>>(y, Wo, out, bo);
}